// HAWPBase_36103495090574
// MI455X (gfx1250) — compile-verified
//
#include <hip/hip_runtime.h>
#include <hip/hip_bf16.h>
#include <stdint.h>

// ---------------------------------------------------------------------------
// HAWP decode for MI455X / gfx1250.
//   H=W=128, C=16, NUM_RESIDUALS=2 -> 5*16384 = 81920 lines
//   loi: 81920 x 512 fp32 (168 MB) ; junctions: 300x2 ; scores: 300
// Strategy:
//   * Kernel 1 (loi): gridDim.y = 4 channel-groups of 4 channels. Each
//     workgroup async-copies its 4 feature planes (256 KB) into LDS via
//     gfx1250 global_load_async_to_lds_b32, interleaved [pixel][c0..c3] so
//     every bilinear corner is ONE ds_load_b128. One wave per line,
//     lane == t-sample (wave32 == 32 sample points).
//   * Kernel 2 (topk): NMS, stable compaction of positive survivors via a
//     blocked LDS prefix scan, bitonic sort of only the positives (padded to
//     pow2), zero-valued tail taken in index order (= exact jax.lax.top_k
//     semantics: descending value, ascending index on ties).
// ---------------------------------------------------------------------------

#define HW      16384      // 128*128
#define NLINES  81920      // 5*HW
#define LPW     320        // lines per workgroup (per channel group)
#define NCHUNK  256        // NCHUNK*LPW == NLINES
#define LOI_ELEMS ((size_t)NLINES * 512)
#define TOPK    300

__device__ __forceinline__ void async_g2lds_b32(uint32_t lds_byte_addr, const void* gaddr) {
    asm volatile("global_load_async_to_lds_b32 %0, %1, off"
                 :: "v"(lds_byte_addr), "v"((unsigned long long)(uintptr_t)gaddr)
                 : "memory");
}

__device__ __forceinline__ void wait_asynccnt0() {
#if __has_builtin(__builtin_amdgcn_s_wait_asynccnt)
    __builtin_amdgcn_s_wait_asynccnt(0);
#else
    asm volatile("s_wait_asynccnt 0" ::: "memory");
#endif
}

__device__ __forceinline__ float clampf(float v, float lo, float hi) {
    return fminf(fmaxf(v, lo), hi);
}

__global__ __launch_bounds__(1024)
void hawp_loi_kernel(const float* __restrict__ md,   // (3,H,W)
                     const float* __restrict__ dis,  // (H,W)
                     const float* __restrict__ res,  // (H,W)
                     const float* __restrict__ feat, // (16,H,W)
                     float* __restrict__ out)        // loi at offset 0
{
    __shared__ float4 tile[HW];   // 256 KB: [pixel][4 channels of this group]

    const int cg  = blockIdx.y;          // channel group 0..3
    const int tid = threadIdx.x;

    // ---- async fill: 4 planes, interleaved per pixel -----------------------
    const float*  fbase   = feat + (size_t)cg * 4 * HW;
    const uint32_t ldsbase = (uint32_t)(uintptr_t)&tile[0];
    for (int e = tid; e < 4 * HW; e += 1024) {
        const int c = e >> 14;           // e / HW  -> channel within group
        const int p = e & (HW - 1);      // pixel   (consecutive lanes: coalesced)
        async_g2lds_b32(ldsbase + (uint32_t)(((p << 2) + c) << 2),
                        fbase + (size_t)c * HW + p);
    }
    wait_asynccnt0();
    __syncthreads();

    // ---- per-line sampling: one wave per line, lane == t index -------------
    const int   wave = tid >> 5;
    const int   lane = tid & 31;
    const float t    = (float)lane * (1.0f / 31.0f);  // linspace(0,1,32)
    const float omt  = 1.0f - t;

    for (int li = wave; li < LPW; li += 32) {
        const int n = blockIdx.x * LPW + li;       // line id = r*HW + y*W + x
        const int r = n >> 14;
        const int p = n & (HW - 1);
        const float y0f = (float)(p >> 7);
        const float x0f = (float)(p & 127);

        // hafm_decoding
        float d = dis[p] + res[p] * (float)(r - 2);
        d = clampf(d, 0.0f, 1.0f);
        const float m0 = md[p], m1 = md[HW + p], m2 = md[2 * HW + p];
        const float mdun = (m0 - 0.5f) * 6.28318530717958647692f;
        const float cs = cosf(mdun);
        const float ss = sinf(mdun);
        const float yst = tanf( m1 * 1.57079632679489661923f);
        const float yed = tanf(-m2 * 1.57079632679489661923f);
        const float ds5 = d * 5.0f;
        const float xs = clampf((cs - ss * yst) * ds5 + x0f, 0.0f, 127.0f);
        const float ys = clampf((ss + cs * yst) * ds5 + y0f, 0.0f, 127.0f);
        const float xe = clampf((cs - ss * yed) * ds5 + x0f, 0.0f, 127.0f);
        const float ye = clampf((ss + cs * yed) * ds5 + y0f, 0.0f, 127.0f);

        // sample point (U*t + V*(1-t) - 0.5), U = start, V = end
        const float px = xs * t + xe * omt - 0.5f;
        const float py = ys * t + ye * omt - 0.5f;
        const float px0 = clampf(floorf(px), 0.0f, 127.0f);
        const float py0 = clampf(floorf(py), 0.0f, 127.0f);
        const float px1 = fminf(px0 + 1.0f, 127.0f);
        const float py1 = fminf(py0 + 1.0f, 127.0f);
        const int ix0 = (int)px0, iy0 = (int)py0;
        const int ix1 = (int)px1, iy1 = (int)py1;
        const float wx0 = px - px0, wx1 = px1 - px;
        const float wy0 = py - py0, wy1 = py1 - py;
        const float w00 = wy1 * wx1, w10 = wy0 * wx1;
        const float w01 = wy1 * wx0, w11 = wy0 * wx0;

        // 4 corners x 4 channels: four ds_load_b128
        const float4 f00 = tile[(iy0 << 7) + ix0];
        const float4 f10 = tile[(iy1 << 7) + ix0];
        const float4 f01 = tile[(iy0 << 7) + ix1];
        const float4 f11 = tile[(iy1 << 7) + ix1];

        const float rx = f00.x * w00 + f10.x * w10 + f01.x * w01 + f11.x * w11;
        const float ry = f00.y * w00 + f10.y * w10 + f01.y * w01 + f11.y * w11;
        const float rz = f00.z * w00 + f10.z * w10 + f01.z * w01 + f11.z * w11;
        const float rw = f00.w * w00 + f10.w * w10 + f01.w * w01 + f11.w * w11;

        // out[n, (cg*4+c)*32 + lane] — coalesced 128B per channel per wave
        float* o = out + (size_t)n * 512 + (size_t)cg * 128 + lane;
        o[0]  = rx;
        o[32] = ry;
        o[64] = rz;
        o[96] = rw;
    }
}

// ---------------------------------------------------------------------------
// NMS + exact top_k(300).
//   key = (f32bits(nms_val) << 32) | (16383 - idx): nms vals are >= 0, so fp
//   bits are order-preserving; descending key order == jax.lax.top_k order.
//   Positives are stably compacted and bitonic-sorted (padded to pow2 with
//   key 0, which sorts below every positive). The zero-valued tail, when
//   needed, is simply the suppressed indices in ascending order.
// ---------------------------------------------------------------------------
#define EPT 16   // elements per thread (1024 * 16 == HW)

__device__ __forceinline__ float nms_val(const float* __restrict__ jloc, int i) {
    const int y = i >> 7, x = i & 127;
    const float a = jloc[i];
    float m = a;
    #pragma unroll
    for (int dy = -1; dy <= 1; ++dy) {
        #pragma unroll
        for (int dx = -1; dx <= 1; ++dx) {
            const int yy = y + dy, xx = x + dx;
            if (yy >= 0 && yy < 128 && xx >= 0 && xx < 128)
                m = fmaxf(m, jloc[(yy << 7) + xx]);
        }
    }
    return (a == m) ? a : 0.0f;   // == a * (a == max3x3)
}

__global__ __launch_bounds__(1024)
void hawp_topk_kernel(const float* __restrict__ jloc,  // (H,W)
                      const float* __restrict__ joff,  // (2,H,W)
                      float* __restrict__ out)
{
    __shared__ unsigned long long posbuf[HW];  // 128 KB (worst case all positive)
    __shared__ int cnt[1024];
    __shared__ int zidx[TOPK];

    const int tid = threadIdx.x;

    // ---- phase 1: count positives in my contiguous chunk -------------------
    int myc = 0;
    #pragma unroll
    for (int k = 0; k < EPT; ++k) {
        if (nms_val(jloc, tid * EPT + k) > 0.0f) myc++;
    }
    cnt[tid] = myc;
    __syncthreads();

    // ---- inclusive Hillis-Steele scan of cnt[0..1023] ----------------------
    for (int d = 1; d < 1024; d <<= 1) {
        const int t = (tid >= d) ? cnt[tid - d] : 0;
        __syncthreads();
        cnt[tid] += t;
        __syncthreads();
    }
    const int npos = cnt[1023];
    const int off  = cnt[tid] - myc;       // exclusive prefix (positives before me)

    // ---- phase 2: stable compaction (recompute NMS; jloc is L0-resident) ---
    int po = off;
    int zo = tid * EPT - off;              // zeros before my chunk
    #pragma unroll
    for (int k = 0; k < EPT; ++k) {
        const int i = tid * EPT + k;
        const float v = nms_val(jloc, i);
        if (v > 0.0f) {
            posbuf[po++] = ((unsigned long long)__float_as_uint(v) << 32)
                         | (unsigned int)(HW - 1 - i);
        } else {
            if (zo < TOPK) zidx[zo] = i;   // first 300 zeros, index order
            zo++;
        }
    }
    __syncthreads();

    // ---- pad positives to pow2 with key 0 (sorts last) ---------------------
    int P = 2;
    while (P < npos) P <<= 1;              // uniform across the block
    for (int i = npos + tid; i < P; i += 1024) posbuf[i] = 0ull;
    __syncthreads();

    // ---- bitonic sort posbuf[0..P) descending ------------------------------
    for (int k = 2; k <= P; k <<= 1) {
        for (int j = k >> 1; j > 0; j >>= 1) {
            for (int i = tid; i < P; i += 1024) {
                const int ixj = i ^ j;
                if (ixj > i) {
                    const unsigned long long a = posbuf[i];
                    const unsigned long long b = posbuf[ixj];
                    const bool swap = ((i & k) == 0) ? (a < b) : (a > b);
                    if (swap) { posbuf[i] = b; posbuf[ixj] = a; }
                }
            }
            __syncthreads();
        }
    }

    // ---- emit junctions + scores -------------------------------------------
    if (tid < TOPK) {
        float v; int idx;
        if (tid < npos) {
            const unsigned long long kk = posbuf[tid];
            v   = __uint_as_float((unsigned int)(kk >> 32));
            idx = HW - 1 - (int)(kk & 0xFFFFFFFFull);
        } else {
            v   = 0.0f;
            idx = zidx[tid - npos];        // zeros: ascending index == top_k order
        }
        const float x = (float)(idx & 127) + joff[idx]      + 0.5f;
        const float y = (float)(idx >> 7)  + joff[HW + idx] + 0.5f;
        out[LOI_ELEMS + (size_t)tid * 2 + 0] = x;
        out[LOI_ELEMS + (size_t)tid * 2 + 1] = y;
        out[LOI_ELEMS + 600 + (size_t)tid]   = v;
    }
}

extern "C" void kernel_launch(void* const* d_in, const int* in_sizes, int n_in,
                              void* d_out, int out_size, void* d_ws, size_t ws_size,
                              hipStream_t stream) {
    const float* md   = (const float*)d_in[0];   // (1,3,128,128)
    const float* dis  = (const float*)d_in[1];   // (1,1,128,128)
    const float* res  = (const float*)d_in[2];   // (1,1,128,128)
    const float* feat = (const float*)d_in[3];   // (16,128,128)
    const float* jloc = (const float*)d_in[4];   // (1,128,128)
    const float* joff = (const float*)d_in[5];   // (2,128,128)
    float* out = (float*)d_out;

    dim3 grid(NCHUNK, 4, 1);   // 256 line-chunks x 4 channel groups
    hawp_loi_kernel<<<grid, 1024, 0, stream>>>(md, dis, res, feat, out);
    hawp_topk_kernel<<<1, 1024, 0, stream>>>(jloc, joff, out);
}